// EmptyLoss_39977555591832
// MI455X (gfx1250) — compile-verified
//
#include <hip/hip_runtime.h>
#include <hip/hip_bf16.h>

typedef float v2f __attribute__((ext_vector_type(2)));
typedef float v8f __attribute__((ext_vector_type(8)));

#define EPS_F 1e-07f
#define LN2_F 0.69314718055994530942f

// Wave-wide f32 sum using V_WMMA_F32_16X16X4_F32.
// A = all-ones 16x4, B = (v, 0) per lane -> every lane's value occupies exactly
// one B[k][n] slot, zeros elsewhere. D[m][n] = sum_k B[k][n] (column sums).
// D VGPR0: lanes 0-15 = colsum[0..15], lanes 16-31 = colsum[0..15] (row M=8
// identical to M=0). Four xor-shuffles then sum the 16 column sums.
// Requires EXEC == all ones (callers guarantee full, uniform waves).
__device__ __forceinline__ float wave_reduce_sum_wmma(float v) {
    v2f a; a.x = 1.0f; a.y = 1.0f;   // A[m][k] = 1 for all 16x4 entries
    v2f b; b.x = v;    b.y = 0.0f;   // one live slot per lane in the 4x16 B
    v8f c = {};
    c = __builtin_amdgcn_wmma_f32_16x16x4_f32(
        /*neg_a=*/false, a, /*neg_b=*/false, b,
        /*c_mod=*/(short)0, c, /*reuse_a=*/false, /*reuse_b=*/false);
    float t = c[0];
    t += __shfl_xor(t, 8, 32);
    t += __shfl_xor(t, 4, 32);
    t += __shfl_xor(t, 2, 32);
    t += __shfl_xor(t, 1, 32);
    return t;  // all lanes hold the full wave sum
}

// Kernel 1: per (sample, block-chunk) partial sums of log2(1 - p + EPS) over
// pixels outside the bbox. Grid: (BLOCKS_PER_SAMPLE, B), block: 256 threads.
// 1M pixels/sample, float4 loads -> 262144 quads; every thread runs exactly
// quads/(gridDim.x*blockDim.x) iterations (uniform -> EXEC all-ones at reduce).
__global__ void __launch_bounds__(256)
masked_log_partial_kernel(const float* __restrict__ pred,
                          const float* __restrict__ bboxes,
                          float* __restrict__ partials) {
    const int H = 1024, W = 1024;
    const int b = blockIdx.y;
    const float4* p4 = (const float4*)(pred + (size_t)b * H * W);

    const float bx0 = bboxes[b * 4 + 0];
    const float by0 = bboxes[b * 4 + 1];
    const float bx1 = bboxes[b * 4 + 2];
    const float by1 = bboxes[b * 4 + 3];
    const int x1 = max((int)floorf(bx0 * (float)W), 0);
    const int y1 = max((int)floorf(by0 * (float)H), 0);
    const int x2 = min((int)floorf(bx1 * (float)W), W);
    const int y2 = min((int)floorf(by1 * (float)H), H);

    const int quads  = (H * W) >> 2;                // 262144
    const int stride = gridDim.x * blockDim.x;

    float t = 0.0f;  // sum of log2(arg) over outside pixels
    for (int q = blockIdx.x * blockDim.x + threadIdx.x; q < quads; q += stride) {
        float4 p = p4[q];               // global_load_b128, coalesced stream
        const int y  = q >> 8;          // (q*4) / W
        const int x0 = (q & 255) << 2;  // (q*4) % W
        const bool rowin = (y >= y1) & (y < y2);

        float a0 = (1.0f - p.x) + EPS_F;
        float a1 = (1.0f - p.y) + EPS_F;
        float a2 = (1.0f - p.z) + EPS_F;
        float a3 = (1.0f - p.w) + EPS_F;
        float l0 = __log2f(a0);
        float l1 = __log2f(a1);
        float l2 = __log2f(a2);
        float l3 = __log2f(a3);
        bool in0 = rowin & (x0 + 0 >= x1) & (x0 + 0 < x2);
        bool in1 = rowin & (x0 + 1 >= x1) & (x0 + 1 < x2);
        bool in2 = rowin & (x0 + 2 >= x1) & (x0 + 2 < x2);
        bool in3 = rowin & (x0 + 3 >= x1) & (x0 + 3 < x2);
        t += in0 ? 0.0f : l0;
        t += in1 ? 0.0f : l1;
        t += in2 ? 0.0f : l2;
        t += in3 ? 0.0f : l3;
    }

    // Block reduction: WMMA wave sums -> LDS -> thread 0 (fixed order).
    __shared__ float wsum[8];
    const float w = wave_reduce_sum_wmma(t);
    const int wid  = threadIdx.x >> 5;
    const int lane = threadIdx.x & 31;
    if (lane == 0) wsum[wid] = w;
    __syncthreads();
    if (threadIdx.x == 0) {
        float s = 0.0f;
        #pragma unroll
        for (int i = 0; i < 8; ++i) s += wsum[i];
        partials[(size_t)b * gridDim.x + blockIdx.x] = s;
    }
}

// Kernel 2: one wave. For each sample: sum its partials (fixed order),
// normalize by the analytic outside-pixel count, accumulate, average over B.
__global__ void __launch_bounds__(32)
masked_log_final_kernel(const float* __restrict__ partials,
                        const float* __restrict__ bboxes,
                        float* __restrict__ out,
                        int B, int nPart) {
    const int H = 1024, W = 1024;
    const int lane = threadIdx.x;
    float acc = 0.0f;
    for (int b = 0; b < B; ++b) {
        float s = 0.0f;
        for (int i = lane; i < nPart; i += 32)   // nPart multiple of 32: uniform
            s += partials[(size_t)b * nPart + i];
        s = wave_reduce_sum_wmma(s);
        if (lane == 0) {
            const int x1 = max((int)floorf(bboxes[b * 4 + 0] * (float)W), 0);
            const int y1 = max((int)floorf(bboxes[b * 4 + 1] * (float)H), 0);
            const int x2 = min((int)floorf(bboxes[b * 4 + 2] * (float)W), W);
            const int y2 = min((int)floorf(bboxes[b * 4 + 3] * (float)H), H);
            const long long inside =
                (long long)max(0, x2 - x1) * (long long)max(0, y2 - y1);
            const float cnt = (float)((long long)H * W - inside);
            const float loss_sum = -LN2_F * s;     // log2 -> natural log
            const float per_sample =
                (cnt > 0.0f) ? (loss_sum / fmaxf(cnt, 1.0f)) : 0.0f;
            acc += per_sample;
        }
    }
    if (lane == 0) out[0] = acc / (float)B;
}

extern "C" void kernel_launch(void* const* d_in, const int* in_sizes, int n_in,
                              void* d_out, int out_size, void* d_ws, size_t ws_size,
                              hipStream_t stream) {
    const float* pred   = (const float*)d_in[0];  // (B,1,H,W) fp32
    const float* bboxes = (const float*)d_in[1];  // (B,4) fp32
    float* out = (float*)d_out;
    float* partials = (float*)d_ws;               // B * BLOCKS_PER_SAMPLE floats

    const int B = in_sizes[1] / 4;                // 32
    const int BLOCKS_PER_SAMPLE = 64;             // 64*256 threads/sample, 16 quads/thread

    dim3 grid1(BLOCKS_PER_SAMPLE, B);
    masked_log_partial_kernel<<<grid1, 256, 0, stream>>>(pred, bboxes, partials);
    masked_log_final_kernel<<<1, 32, 0, stream>>>(partials, bboxes, out,
                                                  B, BLOCKS_PER_SAMPLE);
}